// Attention_Module_40192303956760
// MI455X (gfx1250) — compile-verified
//
#include <hip/hip_runtime.h>
#include <math.h>

typedef __attribute__((ext_vector_type(2))) float v2f;
typedef __attribute__((ext_vector_type(8))) float v8f;

#define B_ 32
#define T_ 4096
#define D_ 512
#define TCHUNKS 4                    // pass-2 t-chunks per batch
#define TCHUNK_LEN (T_ / TCHUNKS)    // 1024
#define DTILES 4                     // pass-2 d tiles of 128 columns

// ---------------------------------------------------------------------------
// Pass 1: cai[b,t] = dot(K[b,t,:], q[b,:]*W[:,0]) + bias   (written into the
// ca region of d_out; softmax normalizes it in place later).
// Wave-level: D[i][j] = sum_k A[i][k]*B[k][j] with A = v-chunk broadcast over
// rows, B = 16 t-rows x 4 d-cols of K. acc[0] on lanes 0..15 = cai[t0+lane].
// ---------------------------------------------------------------------------
__global__ __launch_bounds__(256) void cai_kernel(const float* __restrict__ q,
                                                  const float* __restrict__ K,
                                                  const float* __restrict__ W,
                                                  const float* __restrict__ bias,
                                                  float* __restrict__ ca) {
  __shared__ float v[D_];
  const int b   = blockIdx.x >> 5;       // T/128 = 32 tiles per batch
  const int til = blockIdx.x & 31;
  const int tid = threadIdx.x;

  v[tid]       = q[(long)b * D_ + tid]       * W[tid];
  v[tid + 256] = q[(long)b * D_ + tid + 256] * W[tid + 256];
  __syncthreads();

  const int wave = tid >> 5;
  const int lane = tid & 31;
  const int j    = lane & 15;            // output column = t within the tile
  const int hi   = lane >> 4;            // half-wave: K-dims {0,1} vs {2,3}
  const int t0   = til * 128 + wave * 16;
  const float* Krow = K + ((long)b * T_ + (t0 + j)) * D_;

  v8f acc = {};
  #pragma unroll 4
  for (int s = 0; s < D_ / 4; ++s) {
    const int base = 4 * s + 2 * hi;
    v2f a;  a.x = v[base];  a.y = v[base + 1];          // A: broadcast v-chunk
    v2f bb = *(const v2f*)(Krow + base);                // B: contiguous float2
    acc = __builtin_amdgcn_wmma_f32_16x16x4_f32(false, a, false, bb,
                                                (short)0, acc, false, false);
  }
  if (lane < 16)
    ca[(long)b * T_ + t0 + lane] = acc[0] + bias[0];
}

// ---------------------------------------------------------------------------
// Pass 2a: softmax over T, in place on the ca region. One block per batch.
// ---------------------------------------------------------------------------
__global__ __launch_bounds__(256) void softmax_kernel(float* __restrict__ ca) {
  __shared__ float red[256];
  const int b   = blockIdx.x;
  const int tid = threadIdx.x;
  float x[16];
  float m = -INFINITY;
  #pragma unroll
  for (int i = 0; i < 16; ++i) {
    x[i] = ca[(long)b * T_ + tid + i * 256];
    m = fmaxf(m, x[i]);
  }
  red[tid] = m; __syncthreads();
  for (int s = 128; s > 0; s >>= 1) {
    if (tid < s) red[tid] = fmaxf(red[tid], red[tid + s]);
    __syncthreads();
  }
  m = red[0]; __syncthreads();
  float sum = 0.f;
  #pragma unroll
  for (int i = 0; i < 16; ++i) { x[i] = __expf(x[i] - m); sum += x[i]; }
  red[tid] = sum; __syncthreads();
  for (int s = 128; s > 0; s >>= 1) {
    if (tid < s) red[tid] += red[tid + s];
    __syncthreads();
  }
  const float inv = 1.0f / red[0];
  #pragma unroll
  for (int i = 0; i < 16; ++i)
    ca[(long)b * T_ + tid + i * 256] = x[i] * inv;
}

// ---------------------------------------------------------------------------
// Pass 2b: partial c[b, d] = sum over a 1024-row t-chunk of ca[b,t]*K[b,t,d].
// Wave-level: A = ca-chunk broadcast, B[k][j] = K[b, t+k, d0+j] (each B load
// instruction covers two contiguous 64B row segments). acc[0] on lanes 0..15
// is the partial for 16 d-columns.
// ---------------------------------------------------------------------------
__global__ __launch_bounds__(256) void cpart_kernel(const float* __restrict__ K,
                                                    const float* __restrict__ ca,
                                                    float* __restrict__ part) {
  __shared__ float cach[TCHUNK_LEN];
  const int tc  = blockIdx.x & (TCHUNKS - 1);
  const int dt  = (blockIdx.x >> 2) & (DTILES - 1);
  const int b   = blockIdx.x >> 4;
  const int tid = threadIdx.x;
  const int tBase = tc * TCHUNK_LEN;

  #pragma unroll
  for (int i = 0; i < TCHUNK_LEN / 256; ++i)
    cach[tid + i * 256] = ca[(long)b * T_ + tBase + tid + i * 256];
  __syncthreads();

  const int wave = tid >> 5;
  const int lane = tid & 31;
  const int j    = lane & 15;
  const int hi   = lane >> 4;
  const int dcol = dt * 128 + wave * 16 + j;
  const float* Kb = K + (long)b * T_ * D_;

  v8f acc = {};
  #pragma unroll 4
  for (int s = 0; s < TCHUNK_LEN / 4; ++s) {
    const int tk = 4 * s + 2 * hi;
    v2f a;  a.x = cach[tk];  a.y = cach[tk + 1];        // A: broadcast ca
    v2f bb;
    bb.x = Kb[(long)(tBase + tk)     * D_ + dcol];      // B rows 2*hi, 2*hi+1
    bb.y = Kb[(long)(tBase + tk + 1) * D_ + dcol];
    acc = __builtin_amdgcn_wmma_f32_16x16x4_f32(false, a, false, bb,
                                                (short)0, acc, false, false);
  }
  if (lane < 16)
    part[((long)b * TCHUNKS + tc) * D_ + dt * 128 + wave * 16 + lane] = acc[0];
}

// ---------------------------------------------------------------------------
// Pass 2c: deterministic reduction of the 4 t-chunk partials -> c[b,d].
// ---------------------------------------------------------------------------
__global__ __launch_bounds__(256) void creduce_kernel(const float* __restrict__ part,
                                                      float* __restrict__ c) {
  const int i = blockIdx.x * 256 + threadIdx.x;   // i in [0, B*D)
  const int b = i >> 9;                           // D_ = 512
  const int d = i & (D_ - 1);
  float s = 0.f;
  #pragma unroll
  for (int tc = 0; tc < TCHUNKS; ++tc)
    s += part[((long)b * TCHUNKS + tc) * D_ + d];
  c[i] = s;
}

extern "C" void kernel_launch(void* const* d_in, const int* in_sizes, int n_in,
                              void* d_out, int out_size, void* d_ws, size_t ws_size,
                              hipStream_t stream) {
  const float* q    = (const float*)d_in[0];   // [B, D]
  const float* K    = (const float*)d_in[1];   // [B, T, D]
  const float* W    = (const float*)d_in[2];   // [D, 1]
  const float* bias = (const float*)d_in[3];   // [1]

  float* c    = (float*)d_out;                 // [B, D]
  float* ca   = (float*)d_out + (long)B_ * D_; // [B, T] (logits, then softmax in place)
  float* part = (float*)d_ws;                  // [B, TCHUNKS, D] = 256 KB

  cai_kernel    <<<B_ * (T_ / 128),          256, 0, stream>>>(q, K, W, bias, ca);
  softmax_kernel<<<B_,                       256, 0, stream>>>(ca);
  cpart_kernel  <<<B_ * DTILES * TCHUNKS,    256, 0, stream>>>(K, ca, part);
  creduce_kernel<<<(B_ * D_) / 256,          256, 0, stream>>>(part, c);
}